// CompLayer_13134009991285
// MI455X (gfx1250) — compile-verified
//
#include <hip/hip_runtime.h>
#include <stdint.h>

#define H 128  // embedding dim (fixed by reference)

typedef __attribute__((ext_vector_type(2))) float v2f;
typedef __attribute__((ext_vector_type(8))) float v8f;

// Order-preserving encode of float into uint so atomicMax(u32) == float max.
__device__ __forceinline__ unsigned encf(float f) {
    unsigned u = __float_as_uint(f);
    return (u & 0x80000000u) ? ~u : (u | 0x80000000u);
}
__device__ __forceinline__ float decf(unsigned k) {
    unsigned u = (k & 0x80000000u) ? (k ^ 0x80000000u) : ~k;
    return __uint_as_float(u);
}
// encf(-inf) = ~0xFF800000 = 0x007FFFFF
#define ENC_NEG_INF 0x007FFFFFu

// ---------------- init: zero neigh/den, seed max with -inf ----------------
__global__ void init_kernel(float* __restrict__ neigh, float* __restrict__ den,
                            unsigned* __restrict__ mxEnc, int nNodes) {
    int idx = blockIdx.x * blockDim.x + threadIdx.x;
    if (idx < nNodes * H) neigh[idx] = 0.0f;
    if (idx < nNodes) { den[idx] = 0.0f; mxEnc[idx] = ENC_NEG_INF; }
}

// ------------- per-edge score (wave32 per edge) + segment max -------------
__global__ void score_max_kernel(const float* __restrict__ ent, const float* __restrict__ rel,
                                 const int* __restrict__ src, const int* __restrict__ dst,
                                 const int* __restrict__ rid,
                                 float* __restrict__ score, unsigned* __restrict__ mxEnc,
                                 int nEdges) {
    int gid  = blockIdx.x * blockDim.x + threadIdx.x;
    int e    = gid >> 5;
    int lane = gid & 31;
    if (e >= nEdges) return;
    int s = src[e], d = dst[e], r = rid[e];
    float4 a = ((const float4*)(ent + (size_t)s * H))[lane];
    float4 b = ((const float4*)(rel + (size_t)r * H))[lane];
    float4 c = ((const float4*)(ent + (size_t)d * H))[lane];
    float p = a.x * b.x * c.x + a.y * b.y * c.y + a.z * b.z * c.z + a.w * b.w * c.w;
    #pragma unroll
    for (int off = 16; off > 0; off >>= 1) p += __shfl_xor(p, off, 32);
    if (lane == 0) {
        score[e] = p;
        atomicMax(mxEnc + d, encf(p));
    }
}

// ------------- exp(score - max[dst]) in place + segment sum -------------
__global__ void exp_den_kernel(const int* __restrict__ dst, float* __restrict__ score_exp,
                               const unsigned* __restrict__ mxEnc, float* __restrict__ den,
                               int nEdges) {
    int e = blockIdx.x * blockDim.x + threadIdx.x;
    if (e >= nEdges) return;
    int d = dst[e];
    float ex = __expf(score_exp[e] - decf(mxEnc[d]));
    score_exp[e] = ex;
    atomicAdd(den + d, ex);
}

// ------------- weighted scatter-add of comp into neigh -------------
__global__ void aggregate_kernel(const float* __restrict__ ent, const float* __restrict__ rel,
                                 const int* __restrict__ src, const int* __restrict__ dst,
                                 const int* __restrict__ rid,
                                 const float* __restrict__ expE, const float* __restrict__ den,
                                 float* __restrict__ neigh, int nEdges) {
    int gid  = blockIdx.x * blockDim.x + threadIdx.x;
    int e    = gid >> 5;
    int lane = gid & 31;
    if (e >= nEdges) return;
    int s = src[e], d = dst[e], r = rid[e];
    float alpha = expE[e] / fmaxf(den[d], 1e-30f);
    float4 a = ((const float4*)(ent + (size_t)s * H))[lane];
    float4 b = ((const float4*)(rel + (size_t)r * H))[lane];
    float* o = neigh + (size_t)d * H + lane * 4;
    atomicAdd(o + 0, a.x * b.x * alpha);
    atomicAdd(o + 1, a.y * b.y * alpha);
    atomicAdd(o + 2, a.z * b.z * alpha);
    atomicAdd(o + 3, a.w * b.w * alpha);
}

// ------------- out = tanh(neigh @ W) via V_WMMA_F32_16X16X4_F32 -------------
// Block = 256 threads = 8 waves. Block b owns rows [16b, 16b+16); wave w owns
// output columns [16w, 16w+16). K-loop: 32 x wmma_f32_16x16x4_f32.
// Fragment layouts per CDNA5 ISA 7.12.2 (h = lane>>4, rw = lane&15):
//   A 16x4 f32 : v0 = A[m0+rw][k+2h],  v1 = A[m0+rw][k+2h+1]
//   B 4x16 f32 : v0 = W[k+2h][n0+rw],  v1 = W[k+2h+1][n0+rw]
//   C/D 16x16  : vgpr i -> row m0+i+8h, col n0+rw
__global__ __launch_bounds__(256) void gemm_tanh_kernel(const float* __restrict__ A,
                                                        const float* __restrict__ W,
                                                        float* __restrict__ out, int M) {
    __shared__ float Al[16 * H];  // 8 KB A tile
    int m0  = blockIdx.x * 16;
    int tid = threadIdx.x;
    {   // cooperative coalesced stage of the 16x128 A tile
        const float4* s4 = (const float4*)(A + (size_t)m0 * H);
        float4* d4 = (float4*)Al;
        d4[tid]       = s4[tid];
        d4[tid + 256] = s4[tid + 256];
    }
    __syncthreads();

    int wave = tid >> 5, lane = tid & 31;
    int h = lane >> 4, rw = lane & 15;
    int n0 = wave * 16;

    v8f acc = {};
    #pragma unroll
    for (int k = 0; k < H; k += 4) {
        v2f a, b;
        a.x = Al[rw * H + k + 2 * h];
        a.y = Al[rw * H + k + 2 * h + 1];
        b.x = W[(size_t)(k + 2 * h) * H + n0 + rw];
        b.y = W[(size_t)(k + 2 * h + 1) * H + n0 + rw];
        acc = __builtin_amdgcn_wmma_f32_16x16x4_f32(
            /*neg_a=*/false, a, /*neg_b=*/false, b,
            /*c_mod=*/(short)0, acc, /*reuse_a=*/false, /*reuse_b=*/false);
    }

    #pragma unroll
    for (int i = 0; i < 8; ++i) {
        int m = m0 + i + 8 * h;
        if (m < M) out[(size_t)m * H + n0 + rw] = tanhf(acc[i]);
    }
}

static inline size_t alignup(size_t x) { return (x + 255) & ~(size_t)255; }

extern "C" void kernel_launch(void* const* d_in, const int* in_sizes, int n_in,
                              void* d_out, int out_size, void* d_ws, size_t ws_size,
                              hipStream_t stream) {
    const float* ent = (const float*)d_in[0];
    const float* rel = (const float*)d_in[1];
    const float* W   = (const float*)d_in[2];
    const int*   src = (const int*)d_in[3];
    const int*   dst = (const int*)d_in[4];
    const int*   rid = (const int*)d_in[5];
    float* out = (float*)d_out;

    int nNodes = in_sizes[0] / H;   // 50000
    int nEdges = in_sizes[3];       // 600000

    // workspace layout
    char* ws = (char*)d_ws;
    size_t off = 0;
    float* score = (float*)(ws + off); off = alignup(off + (size_t)nEdges * 4);      // score -> exp in place
    float* den   = (float*)(ws + off); off = alignup(off + (size_t)nNodes * 4);
    unsigned* mx = (unsigned*)(ws + off); off = alignup(off + (size_t)nNodes * 4);
    float* neigh = (float*)(ws + off); off = alignup(off + (size_t)nNodes * H * 4);

    const int BS = 256;
    int initBlocks = (nNodes * H + BS - 1) / BS;
    init_kernel<<<initBlocks, BS, 0, stream>>>(neigh, den, mx, nNodes);

    size_t edgeThreads = (size_t)nEdges * 32;                    // one wave32 per edge
    int edgeWaveBlocks = (int)((edgeThreads + BS - 1) / BS);
    score_max_kernel<<<edgeWaveBlocks, BS, 0, stream>>>(ent, rel, src, dst, rid,
                                                        score, mx, nEdges);

    int edgeBlocks = (nEdges + BS - 1) / BS;
    exp_den_kernel<<<edgeBlocks, BS, 0, stream>>>(dst, score, mx, den, nEdges);

    aggregate_kernel<<<edgeWaveBlocks, BS, 0, stream>>>(ent, rel, src, dst, rid,
                                                        score, den, neigh, nEdges);

    int mTiles = (nNodes + 15) / 16;  // 3125
    gemm_tanh_kernel<<<mTiles, BS, 0, stream>>>(neigh, W, out, nNodes);
}